// CapsuleLayer_54786602828407
// MI455X (gfx1250) — compile-verified
//
#include <hip/hip_runtime.h>

typedef __attribute__((ext_vector_type(2))) float v2f;
typedef __attribute__((ext_vector_type(8))) float v8f;

#define B_TOT 128
#define I_TOT 1152
#define N_CAP 32
#define E_DIM 16
#define D_IN  8
#define NSPLIT 32
#define IPW   (I_TOT / NSPLIT)   /* 36 input capsules per workgroup */
#define BTILE 16

union V8 { v8f v; float f[8]; };

// ---------- VALU lane-permute reductions (no LDS pipe, no DScnt waits) ------
__device__ __forceinline__ float xrow(float v, int m, unsigned s0, unsigned s1) {
#if __has_builtin(__builtin_amdgcn_permlane16)
  unsigned u = __float_as_uint(v);
  return __uint_as_float(__builtin_amdgcn_permlane16(u, u, s0, s1, false, false));
#else
  return __shfl_xor(v, m, 32);
#endif
}
__device__ __forceinline__ float x16(float v) {
#if __has_builtin(__builtin_amdgcn_permlanex16)
  unsigned u = __float_as_uint(v);
  return __uint_as_float(
      __builtin_amdgcn_permlanex16(u, u, 0x76543210u, 0xFEDCBA98u, false, false));
#else
  return __shfl_xor(v, 16, 32);
#endif
}
__device__ __forceinline__ float red16_add(float v) {   // sum within 16-lane rows
  v += xrow(v, 1, 0x67452301u, 0xEFCDAB89u);
  v += xrow(v, 2, 0x54761032u, 0xDCFE98BAu);
  v += xrow(v, 4, 0x32107654u, 0xBA98FEDCu);
  v += xrow(v, 8, 0xFEDCBA98u, 0x76543210u);
  return v;
}
__device__ __forceinline__ float red32_add(float v) { v = red16_add(v); return v + x16(v); }
__device__ __forceinline__ float red32_max(float v) {
  v = fmaxf(v, xrow(v, 1, 0x67452301u, 0xEFCDAB89u));
  v = fmaxf(v, xrow(v, 2, 0x54761032u, 0xDCFE98BAu));
  v = fmaxf(v, xrow(v, 4, 0x32107654u, 0xBA98FEDCu));
  v = fmaxf(v, xrow(v, 8, 0xFEDCBA98u, 0x76543210u));
  return fmaxf(v, x16(v));
}

// LDS overlay: x/v staging is dead by the epilogue, reuse it for the s-transpose.
struct LoopBufs {
  float x[BTILE][IPW][D_IN];      // 18 KB
  float v[BTILE][N_CAP][E_DIM];   // 32 KB
};
union SMem {
  LoopBufs lb;
  float sT[4][8][16][16];         // [wave][tile][e][b], 32 KB
};

// One workgroup = (16-batch tile) x (36 input capsules); 4 waves, wave w owns
// output capsules n = 8w..8w+7. All u_hat tiles are produced TRANSPOSED
// (rows = e, cols = b) by swapping WMMA operands: uhT = W^T x^T. Because A/B
// fragments share the same lane/VGPR striping, the per-lane loads are
// identical to the untransposed version.
__global__ __launch_bounds__(128)
void caps_route_iter(const float* __restrict__ x,      // [B,I,8]
                     const float* __restrict__ Wm,     // [I,32,8,16]
                     const float* __restrict__ bias,   // [I,32]
                     const float* __restrict__ v_prev, // [B,32,16]
                     float* __restrict__ b_ws,         // [B,I,32] routing logits
                     float* __restrict__ partial_s,    // [NSPLIT,B,32,16]
                     int it)
{
  __shared__ SMem sm;
  __shared__ float lds_upd[BTILE][N_CAP];   // 2 KB
  __shared__ float lds_c[BTILE][N_CAP];     // 2 KB

  const int tid  = threadIdx.x;
  const int lane = tid & 31;
  const int wave = tid >> 5;
  const int bl   = lane & 15;    // b-column (and e-column for stores)
  const int hi   = lane >> 4;    // lane half
  const int b0   = blockIdx.y * BTILE;
  const int i0   = blockIdx.x * IPW;
  const int nb   = wave * 8;

  // Stage x[b0:b0+16, i0:i0+IPW, :]
  for (int idx = tid; idx < BTILE * IPW * D_IN; idx += 128) {
    int b = idx / (IPW * D_IN);
    int r = idx % (IPW * D_IN);
    sm.lb.x[b][r / D_IN][r % D_IN] = x[(b0 + b) * (I_TOT * D_IN) + i0 * D_IN + r];
  }
  // Stage previous v
  if (it > 0) {
    for (int idx = tid; idx < BTILE * N_CAP * E_DIM; idx += 128) {
      int b = idx >> 9, r = idx & 511;
      sm.lb.v[b][r >> 4][r & 15] = v_prev[(b0 + b) * (N_CAP * E_DIM) + r];
    }
  }
  __syncthreads();

  V8 sacc[8];                       // transposed s tiles: f[r] = (e=r+8*hi, b=bl)
  #pragma unroll
  for (int t = 0; t < 8; ++t)
    #pragma unroll
    for (int r = 0; r < 8; ++r) sacc[t].f[r] = 0.f;

  for (int ii = 0; ii < IPW; ++ii) {
    const int i = i0 + ii;

    // x^T B-fragments (8d x 16b as two K=4 blocks): VGPR0={K0|K2}, VGPR1={K1|K3}
    v2f bx0, bx1;
    bx0.x = sm.lb.x[bl][ii][2 * hi + 0];
    bx0.y = sm.lb.x[bl][ii][2 * hi + 1];
    bx1.x = sm.lb.x[bl][ii][4 + 2 * hi + 0];
    bx1.y = sm.lb.x[bl][ii][4 + 2 * hi + 1];

    V8 uh[8];                       // transposed u_hat tiles (e-rows, b-cols)
    #pragma unroll
    for (int t = 0; t < 8; ++t) {
      const int n = nb + t;
      const float* wp = Wm + (size_t)(i * N_CAP + n) * (D_IN * E_DIM);
      // W^T A-fragments (16e x 8d): same per-lane values as B-fragments of W
      v2f aw0, aw1;
      aw0.x = wp[(2 * hi + 0) * E_DIM + bl];
      aw0.y = wp[(2 * hi + 1) * E_DIM + bl];
      aw1.x = wp[(4 + 2 * hi + 0) * E_DIM + bl];
      aw1.y = wp[(4 + 2 * hi + 1) * E_DIM + bl];
      v8f acc = {};
      acc = __builtin_amdgcn_wmma_f32_16x16x4_f32(false, aw0, false, bx0,
                                                  (short)0, acc, false, false);
      acc = __builtin_amdgcn_wmma_f32_16x16x4_f32(false, aw1, false, bx1,
                                                  (short)0, acc, false, false);
      uh[t].v = acc;
    }

    // Agreement update: upd[b,n] = sum_e uhT[e,b]*v[b,n,e].
    // Per lane: 8 FMAs over contiguous v, then one cross-half combine.
    if (it > 0) {
      #pragma unroll
      for (int t = 0; t < 8; ++t) {
        const int n = nb + t;
        const float4* vp = (const float4*)&sm.lb.v[bl][n][8 * hi];
        float4 v0 = vp[0], v1 = vp[1];
        float ps = uh[t].f[0] * v0.x + uh[t].f[1] * v0.y +
                   uh[t].f[2] * v0.z + uh[t].f[3] * v0.w +
                   uh[t].f[4] * v1.x + uh[t].f[5] * v1.y +
                   uh[t].f[6] * v1.z + uh[t].f[7] * v1.w;
        float tot = ps + x16(ps);          // both halves now hold upd[b=bl]
        lds_upd[bl][n] = tot;              // 2 lanes/address, same value: safe
      }
    }
    __syncthreads();

    // Softmax over n, wave-parallel: wave w handles rows 4w..4w+3, lane = n.
    #pragma unroll
    for (int q = 0; q < 4; ++q) {
      const int row = wave * 4 + q;
      const float* prevp = (it == 0)
          ? (bias + i * N_CAP)
          : (b_ws + (size_t)(b0 + row) * (I_TOT * N_CAP) + i * N_CAP);
      float l = prevp[lane];
      if (it > 0) l += lds_upd[row][lane];
      float mx = red32_max(l);
      float e  = __expf(l - mx);
      float sm_ = red32_add(e);
      lds_c[row][lane] = e * (1.f / sm_);
      float* bw = b_ws + (size_t)(b0 + row) * (I_TOT * N_CAP) + i * N_CAP;
      bw[lane] = l;                        // updated logits for next iteration
    }
    __syncthreads();

    // s[b,n,e] += c[b,i,n]*u_hat : one c load per tile (lane = b), 8 FMAs.
    #pragma unroll
    for (int t = 0; t < 8; ++t) {
      const int n = nb + t;
      const float cl = lds_c[bl][n];
      #pragma unroll
      for (int r = 0; r < 8; ++r)
        sacc[t].f[r] += cl * uh[t].f[r];
    }
  }

  // Epilogue: transpose s tiles through LDS (overlaid on dead x/v staging),
  // then coalesced partial-s stores.
  __syncthreads();
  #pragma unroll
  for (int t = 0; t < 8; ++t)
    #pragma unroll
    for (int r = 0; r < 8; ++r)
      sm.sT[wave][t][r + 8 * hi][bl] = sacc[t].f[r];     // [e][b]
  #pragma unroll
  for (int t = 0; t < 8; ++t) {
    const int n = nb + t;
    #pragma unroll
    for (int r = 0; r < 8; ++r) {
      const int brow = r + 8 * hi;
      float val = sm.sT[wave][t][bl][brow];              // (b=brow, e=bl)
      partial_s[((blockIdx.x * B_TOT + b0 + brow) * N_CAP + n) * E_DIM + bl] = val;
    }
  }
}

// Reduce partial s over the i-splits and apply squash.
__global__ __launch_bounds__(256)
void caps_squash(const float* __restrict__ partial_s,
                 float* __restrict__ v_ws,
                 float* __restrict__ out,
                 int write_out)
{
  int gid = blockIdx.x * 256 + threadIdx.x;   // 128*32*16 = 65536 total
  float s = 0.f;
  #pragma unroll
  for (int k = 0; k < NSPLIT; ++k)
    s += partial_s[k * (B_TOT * N_CAP * E_DIM) + gid];
  float sq = red16_add(s * s);                // sum over e within 16-lane rows
  float scale = (sq / (1.f + sq)) * rsqrtf(sq);
  float v = scale * s;
  v_ws[gid] = v;
  if (write_out) out[gid] = v;
}

extern "C" void kernel_launch(void* const* d_in, const int* in_sizes, int n_in,
                              void* d_out, int out_size, void* d_ws, size_t ws_size,
                              hipStream_t stream) {
  (void)in_sizes; (void)n_in; (void)out_size; (void)ws_size;
  const float* x    = (const float*)d_in[0];
  const float* Wm   = (const float*)d_in[1];
  const float* bias = (const float*)d_in[2];
  float* out = (float*)d_out;

  char* ws = (char*)d_ws;
  const size_t bws_bytes = (size_t)B_TOT * I_TOT * N_CAP * 4;   // 18.9 MB
  const size_t v_bytes   = (size_t)B_TOT * N_CAP * E_DIM * 4;   // 256 KB
  float* b_ws = (float*)ws;
  float* v_ws = (float*)(ws + bws_bytes);
  float* ps   = (float*)(ws + bws_bytes + v_bytes);             // 8.4 MB

  dim3 grid1(NSPLIT, B_TOT / BTILE);   // 32 x 8 = 256 workgroups
  for (int it = 0; it < 3; ++it) {
    caps_route_iter<<<grid1, dim3(128), 0, stream>>>(x, Wm, bias, v_ws, b_ws, ps, it);
    caps_squash<<<(B_TOT * N_CAP * E_DIM) / 256, dim3(256), 0, stream>>>(
        ps, v_ws, out, it == 2 ? 1 : 0);
  }
}